// GCNModel_4080218931695
// MI455X (gfx1250) — compile-verified
//
#include <hip/hip_runtime.h>

#define NN 100000

typedef float v2f __attribute__((ext_vector_type(2)));
typedef float v8f __attribute__((ext_vector_type(8)));

// ---------------- degree / normalization ----------------
__global__ void deg_kernel(const int* __restrict__ dst, float* __restrict__ deg, int E) {
    int i = blockIdx.x * blockDim.x + threadIdx.x;
    if (i < E) atomicAdd(&deg[dst[i]], 1.0f);
}

__global__ void isd_kernel(const float* __restrict__ deg, float* __restrict__ isd, int n) {
    int i = blockIdx.x * blockDim.x + threadIdx.x;
    if (i < n) {
        float d = deg[i];
        isd[i] = (d > 0.0f) ? rsqrtf(fmaxf(d, 1.0f)) : 0.0f;
    }
}

// ---------------- layer-1 GEMM: hs1 = rowscale(isd) * (X @ W1) ----------------
// X [N,256], W1 [256,128] row-major, hs1 [N,128]. One wave -> 16x128 strip.
__global__ __launch_bounds__(256, 2) void gemm1_kernel(
    const float* __restrict__ X, const float* __restrict__ W1,
    const float* __restrict__ isd, float* __restrict__ hs1, int nTiles) {
    __shared__ float w1t[128 * 256];  // W1 transposed: w1t[c*256+k] = W1[k*128+c]  (128 KB)
    for (int i = threadIdx.x; i < 256 * 128; i += blockDim.x) {
        int k = i >> 7, c = i & 127;
        w1t[c * 256 + k] = W1[i];
    }
    __syncthreads();
    int wave = threadIdx.x >> 5, lane = threadIdx.x & 31;
    int tile = blockIdx.x * 8 + wave;
    if (tile >= nTiles) return;
    int row0 = tile * 16;
    int m = lane & 15, kh = lane >> 4;                 // A: lane holds M=m, K base 2*kh
    const float* xrow = X + (size_t)(row0 + m) * 256 + 2 * kh;
    v8f acc[8] = {};
    for (int kk = 0; kk < 256; kk += 4) {
        v2f a = *(const v2f*)(xrow + kk);
        #pragma unroll
        for (int c = 0; c < 8; ++c) {
            v2f b = *(const v2f*)(&w1t[(c * 16 + m) * 256 + kk + 2 * kh]);
            acc[c] = __builtin_amdgcn_wmma_f32_16x16x4_f32(
                false, a, false, b, (short)0, acc[c], false, false);
        }
    }
    #pragma unroll
    for (int r = 0; r < 8; ++r) {
        int row = row0 + r + 8 * kh;                   // C/D: vgpr r -> M = r + 8*(lane>=16)
        float s = isd[row];
        float* o = hs1 + (size_t)row * 128 + m;
        #pragma unroll
        for (int c = 0; c < 8; ++c) o[c * 16] = s * acc[c][r];
    }
}

// ---------------- edge scatter, layer 1 (128 floats/edge, one wave/edge) ----------------
__global__ void scatter1_kernel(const int* __restrict__ src, const int* __restrict__ dst,
                                const float* __restrict__ hs1, float* __restrict__ acc1, int E) {
    int w = (blockIdx.x * blockDim.x + threadIdx.x) >> 5;
    int lane = threadIdx.x & 31;
    if (w >= E) return;
    int s = src[w], d = dst[w];
    float4 v = ((const float4*)(hs1 + (size_t)s * 128))[lane];
    float* pd = acc1 + (size_t)d * 128 + lane * 4;
    atomicAdd(pd + 0, v.x);
    atomicAdd(pd + 1, v.y);
    atomicAdd(pd + 2, v.z);
    atomicAdd(pd + 3, v.w);
}

// ---------------- t = relu(isd[n]*acc1 + b1)  (written over hs1) ----------------
__global__ void finish1_kernel(const float* __restrict__ acc1, const float* __restrict__ isd,
                               const float* __restrict__ b1, float* __restrict__ t, int total) {
    int i = blockIdx.x * blockDim.x + threadIdx.x;
    if (i >= total) return;
    int n = i >> 7, j = i & 127;
    float v = fmaf(isd[n], acc1[i], b1[j]);
    t[i] = v > 0.0f ? v : 0.0f;
}

// ---------------- layer-2 GEMM: hs2 = rowscale(isd) * (T @ W2), padded to 48 cols --------
__global__ __launch_bounds__(256, 2) void gemm2_kernel(
    const float* __restrict__ T, const float* __restrict__ W2,
    const float* __restrict__ isd, float* __restrict__ hs2, int nTiles) {
    __shared__ float w2t[48 * 128];   // transposed + zero-padded: w2t[c*128+k]  (24 KB)
    for (int i = threadIdx.x; i < 48 * 128; i += blockDim.x) {
        int c = i >> 7, k = i & 127;
        w2t[i] = (c < 40) ? W2[k * 40 + c] : 0.0f;
    }
    __syncthreads();
    int wave = threadIdx.x >> 5, lane = threadIdx.x & 31;
    int tile = blockIdx.x * 8 + wave;
    if (tile >= nTiles) return;
    int row0 = tile * 16;
    int m = lane & 15, kh = lane >> 4;
    const float* trow = T + (size_t)(row0 + m) * 128 + 2 * kh;
    v8f acc[3] = {};
    for (int kk = 0; kk < 128; kk += 4) {
        v2f a = *(const v2f*)(trow + kk);
        #pragma unroll
        for (int c = 0; c < 3; ++c) {
            v2f b = *(const v2f*)(&w2t[(c * 16 + m) * 128 + kk + 2 * kh]);
            acc[c] = __builtin_amdgcn_wmma_f32_16x16x4_f32(
                false, a, false, b, (short)0, acc[c], false, false);
        }
    }
    #pragma unroll
    for (int r = 0; r < 8; ++r) {
        int row = row0 + r + 8 * kh;
        float s = isd[row];
        float* o = hs2 + (size_t)row * 48 + m;
        #pragma unroll
        for (int c = 0; c < 3; ++c) o[c * 16] = s * acc[c][r];
    }
}

// ---------------- edge scatter, layer 2 (40 floats/edge, one wave/edge) ----------------
__global__ void scatter2_kernel(const int* __restrict__ src, const int* __restrict__ dst,
                                const float* __restrict__ hs2, float* __restrict__ acc2, int E) {
    int w = (blockIdx.x * blockDim.x + threadIdx.x) >> 5;
    int lane = threadIdx.x & 31;
    if (w >= E) return;
    int s = src[w], d = dst[w];
    const float* ps = hs2 + (size_t)s * 48;
    float* pd = acc2 + (size_t)d * 40;
    atomicAdd(pd + lane, ps[lane]);
    if (lane < 8) atomicAdd(pd + 32 + lane, ps[32 + lane]);
}

// ---------------- out = isd[n]*acc2 + b2 ----------------
__global__ void finish2_kernel(const float* __restrict__ acc2, const float* __restrict__ isd,
                               const float* __restrict__ b2, float* __restrict__ out, int total) {
    int i = blockIdx.x * blockDim.x + threadIdx.x;
    if (i >= total) return;
    int n = i / 40, j = i - n * 40;
    out[i] = fmaf(isd[n], acc2[i], b2[j]);
}

extern "C" void kernel_launch(void* const* d_in, const int* in_sizes, int n_in,
                              void* d_out, int out_size, void* d_ws, size_t ws_size,
                              hipStream_t stream) {
    const float* x  = (const float*)d_in[0];
    const int*   src = (const int*)d_in[1];
    const int*   dst = (const int*)d_in[2];
    const float* W1 = (const float*)d_in[3];
    const float* b1 = (const float*)d_in[4];
    const float* W2 = (const float*)d_in[5];
    const float* b2 = (const float*)d_in[6];
    float* out = (float*)d_out;

    const int N = NN;
    const int E = in_sizes[1];

    // workspace layout (floats): deg N | isd N | hs1 128N | acc1 128N | hs2 48N | acc2 40N
    float* ws   = (float*)d_ws;
    float* deg  = ws;
    float* isd  = deg + N;
    float* hs1  = isd + N;
    float* acc1 = hs1 + (size_t)128 * N;
    float* hs2  = acc1 + (size_t)128 * N;
    float* acc2 = hs2 + (size_t)48 * N;

    hipMemsetAsync(deg,  0, sizeof(float) * (size_t)N,        stream);
    hipMemsetAsync(acc1, 0, sizeof(float) * (size_t)128 * N,  stream);
    hipMemsetAsync(acc2, 0, sizeof(float) * (size_t)40 * N,   stream);

    deg_kernel<<<(E + 255) / 256, 256, 0, stream>>>(dst, deg, E);
    isd_kernel<<<(N + 255) / 256, 256, 0, stream>>>(deg, isd, N);

    const int nTiles = N / 16;  // 6250 (N divisible by 16)
    gemm1_kernel<<<(nTiles + 7) / 8, 256, 0, stream>>>(x, W1, isd, hs1, nTiles);
    scatter1_kernel<<<(E + 7) / 8, 256, 0, stream>>>(src, dst, hs1, acc1, E);
    finish1_kernel<<<(N * 128 + 255) / 256, 256, 0, stream>>>(acc1, isd, b1, hs1, N * 128);
    gemm2_kernel<<<(nTiles + 7) / 8, 256, 0, stream>>>(hs1, W2, isd, hs2, nTiles);
    scatter2_kernel<<<(E + 7) / 8, 256, 0, stream>>>(src, dst, hs2, acc2, E);
    finish2_kernel<<<(N * 40 + 255) / 256, 256, 0, stream>>>(acc2, isd, b2, out, N * 40);
}